// RGBlock_91182155694147
// MI455X (gfx1250) — compile-verified
//
#include <hip/hip_runtime.h>
#include <stdint.h>
#include <math.h>

#define HD 64  // feature width (IN == H == 64)

typedef __attribute__((ext_vector_type(2))) float v2f;
typedef __attribute__((ext_vector_type(8))) float v8f;

#if defined(__has_builtin)
#if __has_builtin(__builtin_amdgcn_wmma_f32_16x16x4_f32)
#define HAVE_WMMA_F32 1
#endif
#endif

// ---------- helpers: monotone float<->uint map for atomicMax-based segment max ----------
static __device__ __forceinline__ unsigned mapf(float f) {
  unsigned b = __float_as_uint(f);
  return (b & 0x80000000u) ? ~b : (b | 0x80000000u);
}
static __device__ __forceinline__ float unmapf(unsigned u) {
  unsigned b = (u & 0x80000000u) ? (u & 0x7FFFFFFFu) : ~u;
  return __uint_as_float(b);
}

// ---------- trivial fills ----------
__global__ void fill_u32_kernel(unsigned* __restrict__ p, unsigned v, int n) {
  int i = blockIdx.x * blockDim.x + threadIdx.x;
  if (i < n) p[i] = v;
}

// ---------- degree: deg[dst] += 1 (deg pre-filled with 1.0 for the self loop) ----------
__global__ void deg_kernel(const int* __restrict__ dst, float* __restrict__ deg, int e) {
  int i = blockIdx.x * blockDim.x + threadIdx.x;
  if (i < e) unsafeAtomicAdd(&deg[dst[i]], 1.0f);
}

__global__ void dinv_kernel(const float* __restrict__ deg, float* __restrict__ dinv, int n) {
  int i = blockIdx.x * blockDim.x + threadIdx.x;
  if (i < n) dinv[i] = rsqrtf(deg[i]);
}

// ---------- GEMM: Y[n,64] = X[n,64] @ W[64,64] via V_WMMA_F32_16X16X4_F32 ----------
// 128 threads = 4 waves; each wave owns a 16-row x 64-col output tile.
__global__ __launch_bounds__(128) void gemm64_wmma_kernel(const float* __restrict__ X,
                                                          const float* __restrict__ W,
                                                          float* __restrict__ Y, int n) {
  __shared__ float wlds[HD * HD];
  for (int i = threadIdx.x; i < HD * HD; i += 128) wlds[i] = W[i];
  __syncthreads();

#if defined(HAVE_WMMA_F32)
  const int wave = threadIdx.x >> 5;
  const int lane = threadIdx.x & 31;
  const int row0 = (blockIdx.x * 4 + wave) * 16;
  const int rowTiles = (n + 15) >> 4;
  if ((blockIdx.x * 4 + wave) >= rowTiles) return;  // wave-uniform: EXEC stays all-ones

  const int mrow  = lane & 15;  // A-row within tile; also B/D column index
  const int khalf = lane >> 4;  // 0 -> K{0,1}, 1 -> K{2,3}

  // clamp row for loads on a partial last tile (duplicate reads are harmless)
  int rload = row0 + mrow;
  if (rload > n - 1) rload = n - 1;
  const float* __restrict__ xrow = X + (size_t)rload * HD;

  v8f acc[4] = {v8f{}, v8f{}, v8f{}, v8f{}};

  for (int k0 = 0; k0 < HD; k0 += 4) {
    const int ka = k0 + khalf * 2;
    v2f a;
    a.x = xrow[ka];
    a.y = xrow[ka + 1];
#pragma unroll
    for (int t = 0; t < 4; ++t) {
      const int col = t * 16 + mrow;
      v2f b;
      b.x = wlds[ka * HD + col];
      b.y = wlds[(ka + 1) * HD + col];
      acc[t] = __builtin_amdgcn_wmma_f32_16x16x4_f32(false, a, false, b,
                                                     (short)0, acc[t], false, false);
    }
  }

#pragma unroll
  for (int t = 0; t < 4; ++t) {
    const int col = t * 16 + mrow;
#pragma unroll
    for (int j = 0; j < 8; ++j) {
      const int r = row0 + j + khalf * 8;  // D layout: vgpr j -> rows j / j+8
      if (r < n) Y[(size_t)r * HD + col] = acc[t][j];
    }
  }
#else
  // scalar fallback (should not be used; histogram check drives iteration)
  for (int i = blockIdx.x * blockDim.x + threadIdx.x; i < n * HD;
       i += gridDim.x * blockDim.x) {
    int r = i >> 6, c = i & 63;
    float s = 0.f;
    for (int k = 0; k < HD; ++k) s += X[(size_t)r * HD + k] * wlds[k * HD + c];
    Y[i] = s;
  }
#endif
}

// ---------- edge aggregation: agg[dst] += h[src] * dinv[src]*dinv[dst] ----------
// one thread = (edge, 4 columns)
__global__ void agg_kernel(const float* __restrict__ h, const int* __restrict__ src,
                           const int* __restrict__ dst, const float* __restrict__ dinv,
                           float* __restrict__ agg, int e) {
  int i = blockIdx.x * blockDim.x + threadIdx.x;
  if (i >= e * 16) return;
  const int ed = i >> 4;
  const int cg = (i & 15) << 2;
  const int s = src[ed], d = dst[ed];
  const float nrm = dinv[s] * dinv[d];
  const float4 v = *(const float4*)(h + (size_t)s * HD + cg);
  float* ap = agg + (size_t)d * HD + cg;
  unsafeAtomicAdd(ap + 0, v.x * nrm);
  unsafeAtomicAdd(ap + 1, v.y * nrm);
  unsafeAtomicAdd(ap + 2, v.z * nrm);
  unsafeAtomicAdd(ap + 3, v.w * nrm);
}

// ---------- out = relu(agg + xw/deg + bias) ----------
__global__ void combine_kernel(const float* __restrict__ agg, const float* __restrict__ xw,
                               const float* __restrict__ deg, const float* __restrict__ bias,
                               float* __restrict__ out, int n) {
  int i = blockIdx.x * blockDim.x + threadIdx.x;
  if (i < n * HD) {
    const int r = i >> 6, c = i & 63;
    float v = agg[i] + xw[i] / deg[r] + bias[c];
    out[i] = v > 0.f ? v : 0.f;
  }
}

// ---------- segment max over clusters (mapped-uint atomicMax) ----------
__global__ void pool_max_kernel(const float* __restrict__ h, const int* __restrict__ cluster,
                                unsigned* __restrict__ outm, int n) {
  int i = blockIdx.x * blockDim.x + threadIdx.x;
  if (i < n * HD) {
    const int r = i >> 6, c = i & 63;
    atomicMax(&outm[(size_t)cluster[r] * HD + c], mapf(h[i]));
  }
}

__global__ void unmap_kernel(unsigned* __restrict__ outm, int n) {
  int i = blockIdx.x * blockDim.x + threadIdx.x;
  if (i < n) {
    const unsigned u = outm[i];
    const float f = (u == 0u) ? 0.0f : unmapf(u);  // untouched cluster slot -> 0.0
    outm[i] = __float_as_uint(f);
  }
}

// ---------- coarse edges: out[i] = (float)cluster[edge[i]] ----------
__global__ void edge_relabel_kernel(const int* __restrict__ edges, const int* __restrict__ cluster,
                                    float* __restrict__ out, int m) {
  int i = blockIdx.x * blockDim.x + threadIdx.x;
  if (i < m) out[i] = (float)cluster[edges[i]];
}

// ---------- batch_coarse: segment min ----------
__global__ void batch_min_kernel(const int* __restrict__ batch, const int* __restrict__ cluster,
                                 int* __restrict__ bmin, int n) {
  int i = blockIdx.x * blockDim.x + threadIdx.x;
  if (i < n) atomicMin(&bmin[cluster[i]], batch[i]);
}

__global__ void batch_finalize_kernel(const int* __restrict__ bmin, float* __restrict__ out, int n) {
  int i = blockIdx.x * blockDim.x + threadIdx.x;
  if (i < n) out[i] = (bmin[i] == 0x7FFFFFFF) ? 0.0f : (float)bmin[i];
}

// =======================================================================================
extern "C" void kernel_launch(void* const* d_in, const int* in_sizes, int n_in,
                              void* d_out, int out_size, void* d_ws, size_t ws_size,
                              hipStream_t stream) {
  const float* x   = (const float*)d_in[0];
  const float* W1  = (const float*)d_in[1];
  const float* b1  = (const float*)d_in[2];
  const float* W2  = (const float*)d_in[3];
  const float* b2  = (const float*)d_in[4];
  const int* edges = (const int*)d_in[5];
  const int* clus  = (const int*)d_in[6];
  const int* batch = (const int*)d_in[7];

  const int n = in_sizes[0] / HD;   // 100000 nodes
  const int e = in_sizes[5] / 2;    // 1200000 edges
  const int* src = edges;
  const int* dst = edges + e;

  // workspace layout (all 4-byte slots)
  float* deg  = (float*)d_ws;
  float* dinv = deg + n;
  float* bufA = dinv + n;                  // xw / h1 / h2
  float* bufB = bufA + (size_t)n * HD;     // agg1, then h1@W2
  float* bufC = bufB + (size_t)n * HD;     // agg2
  int*   bmin = (int*)(bufC + (size_t)n * HD);

  float* out_x     = (float*)d_out;                 // [n, 64]
  float* out_edge  = out_x + (size_t)n * HD;        // [2, e]
  float* out_batch = out_edge + (size_t)2 * e;      // [n]

  const int TB = 256;
  const int nh = n * HD;
  const int gNH = (nh + TB - 1) / TB;
  const int gN  = (n + TB - 1) / TB;
  const int gE  = (e + TB - 1) / TB;
  const int gE16 = (e * 16 + TB - 1) / TB;
  const int rowTiles = (n + 15) >> 4;
  const int gGemm = (rowTiles + 3) / 4;

  const unsigned ONE_F32_BITS = 0x3F800000u;  // host-side bit pattern of 1.0f

  // ---- init ----
  fill_u32_kernel<<<gN, TB, 0, stream>>>((unsigned*)deg, ONE_F32_BITS, n);  // self loop
  fill_u32_kernel<<<gNH, TB, 0, stream>>>((unsigned*)bufB, 0u, nh);         // agg1 = 0
  fill_u32_kernel<<<gNH, TB, 0, stream>>>((unsigned*)bufC, 0u, nh);         // agg2 = 0
  fill_u32_kernel<<<gNH, TB, 0, stream>>>((unsigned*)out_x, 0u, nh);        // mapped -max sentinel
  fill_u32_kernel<<<gN, TB, 0, stream>>>((unsigned*)bmin, 0x7FFFFFFFu, n);  // INT_MAX

  // ---- degree / norm ----
  deg_kernel<<<gE, TB, 0, stream>>>(dst, deg, e);
  dinv_kernel<<<gN, TB, 0, stream>>>(deg, dinv, n);

  // ---- layer 1: h1 = relu(agg(x@W1) + (x@W1)/deg + b1) ----
  gemm64_wmma_kernel<<<gGemm, 128, 0, stream>>>(x, W1, bufA, n);
  agg_kernel<<<gE16, TB, 0, stream>>>(bufA, src, dst, dinv, bufB, e);
  combine_kernel<<<gNH, TB, 0, stream>>>(bufB, bufA, deg, b1, bufA, n);

  // ---- layer 2: h2 = relu(agg(h1) + (h1@W2)/deg + b2) ----
  gemm64_wmma_kernel<<<gGemm, 128, 0, stream>>>(bufA, W2, bufB, n);
  agg_kernel<<<gE16, TB, 0, stream>>>(bufA, src, dst, dinv, bufC, e);
  combine_kernel<<<gNH, TB, 0, stream>>>(bufC, bufB, deg, b2, bufA, n);

  // ---- max pool over clusters ----
  pool_max_kernel<<<gNH, TB, 0, stream>>>(bufA, clus, (unsigned*)out_x, n);
  unmap_kernel<<<gNH, TB, 0, stream>>>((unsigned*)out_x, nh);

  // ---- coarse edges + coarse batch ----
  edge_relabel_kernel<<<(2 * e + TB - 1) / TB, TB, 0, stream>>>(edges, clus, out_edge, 2 * e);
  batch_min_kernel<<<gN, TB, 0, stream>>>(batch, clus, bmin, n);
  batch_finalize_kernel<<<gN, TB, 0, stream>>>(bmin, out_batch, n);
}